// GINConv_23484881175228
// MI455X (gfx1250) — compile-verified
//
#include <hip/hip_runtime.h>

// TCGNN fused forward for MI455X (gfx1250, wave32):
//   out = segment_sum(X[col]) @ W  ==  A * X * W, fused per 16-row window.
// Phase 1: CSR neighbor aggregation -> dense 16x64 fp32 tile in LDS.
// Phase 2: tile(16x64) @ W(64x64) via V_WMMA_F32_16X16X4_F32 (full fp32 precision).

#define D_IN   64
#define D_OUT  64
#define RW     16      // row-window (rows per block)
#define PITCH  68      // LDS pitch in floats: bank = (4*mr + k) % 64 -> conflict-free A reads

typedef __attribute__((ext_vector_type(2))) float v2f;
typedef __attribute__((ext_vector_type(8))) float v8f;

__global__ __launch_bounds__(128)
void tcgnn_fused_kernel(const float* __restrict__ X,
                        const float* __restrict__ W,
                        const int*   __restrict__ rowptr,
                        const int*   __restrict__ colidx,
                        float*       __restrict__ out,
                        int n_nodes)
{
    __shared__ float tile[RW][PITCH];

    const int tid  = threadIdx.x;
    const int wave = tid >> 5;   // 0..3
    const int lane = tid & 31;
    const int r0   = blockIdx.x * RW;

    // ---------------- Phase 1: CSR segment-sum into LDS tile ----------------
    // Wave w handles rows w, w+4, w+8, w+12 of the window. Each lane owns two
    // feature columns (2*lane, 2*lane+1) -> one float2 (global_load_b64) per
    // neighbor per lane; a full wave pulls the 256B neighbor row coalesced.
    for (int rr = wave; rr < RW; rr += 4) {
        const int row = r0 + rr;
        float ax = 0.0f, ay = 0.0f;
        if (row < n_nodes) {                  // wave-uniform guard
            const int e_beg = rowptr[row];
            const int e_end = rowptr[row + 1];
            int c_next = (e_beg < e_end) ? colidx[e_beg] : 0;
            for (int e = e_beg; e < e_end; ++e) {
                const int c = c_next;
                if (e + 1 < e_end) {
                    c_next = colidx[e + 1];
                    // Hide L2 latency of the dependent colidx -> X chain.
                    __builtin_prefetch(X + (size_t)c_next * D_IN + 2 * lane, 0, 0);
                }
                const float2 v =
                    ((const float2*)(X + (size_t)c * D_IN))[lane];
                ax += v.x;
                ay += v.y;
            }
        }
        tile[rr][2 * lane]     = ax;
        tile[rr][2 * lane + 1] = ay;
    }

    __syncthreads();

    // ---------------- Phase 2: (16x64 tile) @ (64x64 W) via f32 WMMA --------
    // Wave w computes output N-tile n0 = 16*w. K-loop: 16 steps of K=4.
    // A (16x4 f32) layout: lane = {mr, half}; VGPR0 = K=2*half, VGPR1 = K=2*half+1.
    // B (4x16 f32) layout: N = lane%16; VGPR0 = K=2*half, VGPR1 = K=2*half+1.
    // C/D (16x16 f32): VGPR r -> M = r + 8*half, N = lane%16.
    const int n0   = wave << 4;
    const int mr   = lane & 15;
    const int half = lane >> 4;

    v8f acc = {};
    #pragma unroll
    for (int k0 = 0; k0 < D_IN; k0 += 4) {
        const int ka = k0 + 2 * half;
        v2f a, b;
        a[0] = tile[mr][ka];
        a[1] = tile[mr][ka + 1];
        b[0] = W[(size_t)ka * D_OUT + n0 + mr];
        b[1] = W[(size_t)(ka + 1) * D_OUT + n0 + mr];
        // (neg_a, A, neg_b, B, c_mod, C, reuse_a, reuse_b)
        acc = __builtin_amdgcn_wmma_f32_16x16x4_f32(
            false, a, false, b, (short)0, acc, false, false);
    }

    // Epilogue: wave-uniform full-tile fast path -> 8 unguarded coalesced
    // b32 stores (one per C VGPR / output row); guarded path only for a
    // ragged final window (never taken when n_nodes % 16 == 0).
    float* const dst = out + (size_t)(r0 + half * 8) * D_OUT + n0 + mr;
    if (r0 + RW <= n_nodes) {
        #pragma unroll
        for (int r = 0; r < 8; ++r) {
            dst[(size_t)r * D_OUT] = acc[r];
        }
    } else {
        #pragma unroll
        for (int r = 0; r < 8; ++r) {
            if (r0 + r + half * 8 < n_nodes) {
                dst[(size_t)r * D_OUT] = acc[r];
            }
        }
    }
}

extern "C" void kernel_launch(void* const* d_in, const int* in_sizes, int n_in,
                              void* d_out, int out_size, void* d_ws, size_t ws_size,
                              hipStream_t stream) {
    // setup_inputs() order:
    //   0: X [N_NODES*D_IN] f32
    //   1: weights [D_IN*D_OUT] f32
    //   2: row_pointers [N_NODES+1] i32
    //   3: column_index [N_EDGES] i32
    //   4..6: TCGNN metadata (unused by the math)
    const float* X      = (const float*)d_in[0];
    const float* W      = (const float*)d_in[1];
    const int*   rowptr = (const int*)d_in[2];
    const int*   colidx = (const int*)d_in[3];
    float*       out    = (float*)d_out;

    const int n_nodes = in_sizes[2] - 1;   // row_pointers has N_NODES+1 entries

    dim3 grid((n_nodes + RW - 1) / RW);
    dim3 block(128);                        // 4 wave32 waves
    tcgnn_fused_kernel<<<grid, block, 0, stream>>>(X, W, rowptr, colidx, out, n_nodes);
}